// PrimitiveFitting_67997922230537
// MI455X (gfx1250) — compile-verified
//
#include <hip/hip_runtime.h>
#include <hip/hip_bf16.h>
#include <math.h>

// ---------------------------------------------------------------------------
// PrimitiveFitting for gfx1250 (MI455X, wave32)
//  dense-key grid (<=8*47^3 keys) replaces sort/unique/searchsorted
//  per-point 27-stencil rank-1 accumulation via V_WMMA_F32_16X16X4_F32
//  branch-free fp32 atomic scatter (dump row + padded stride 16)
//  async global->LDS staging (gfx1250 ASYNCcnt path) in the scan
// ---------------------------------------------------------------------------

#define K_CAP       (1 << 20)            // dense key-space capacity (real max ~831k)
#define SCAN_TILE   1024
#define SCAN_BLOCKS (K_CAP / SCAN_TILE)  // 1024
#define NCOMP       14                   // [w, w*p4(4), w*outer(xyz)(9)]
#define NSTRIDE     16                   // padded accumulator row stride

typedef __attribute__((ext_vector_type(2))) float v2f;
typedef __attribute__((ext_vector_type(8))) float v8f;
typedef __attribute__((ext_vector_type(4))) int   v4i;

#if defined(__has_builtin)
#if __has_builtin(__builtin_amdgcn_wmma_f32_16x16x4_f32)
#define HAS_WMMA_F32X4 1
#endif
#if __has_builtin(__builtin_amdgcn_global_load_async_to_lds_b128) && \
    __has_builtin(__builtin_amdgcn_s_wait_asynccnt)
#define HAS_ASYNC_LDS 1
#endif
#endif

struct Params {
  float pcmin[4];
  int   dims[4];
  int   keyspace;
  int   numUnique;
};

// order-preserving float <-> uint encoding for atomic min/max
__device__ __forceinline__ unsigned fenc(float f) {
  unsigned u = __float_as_uint(f);
  return (u & 0x80000000u) ? ~u : (u | 0x80000000u);
}
__device__ __forceinline__ float fdec(unsigned e) {
  unsigned u = (e & 0x80000000u) ? (e ^ 0x80000000u) : ~e;
  return __uint_as_float(u);
}

__global__ void k_init(unsigned* gmm) {
  int t = threadIdx.x;
  if (t < 4) gmm[t] = 0xFFFFFFFFu;
  else if (t < 8) gmm[t] = 0u;
}

__global__ void k_minmax(const float* __restrict__ pts, int n, unsigned* gmm) {
  __shared__ unsigned smin[4], smax[4];
  int tid = threadIdx.x;
  if (tid < 4) { smin[tid] = 0xFFFFFFFFu; smax[tid] = 0u; }
  __syncthreads();
  int i = blockIdx.x * blockDim.x + tid;
  if (i < n) {
    float4 p = reinterpret_cast<const float4*>(pts)[i];
    unsigned e0 = fenc(p.x), e1 = fenc(p.y), e2 = fenc(p.z), e3 = fenc(p.w);
    atomicMin(&smin[0], e0); atomicMax(&smax[0], e0);
    atomicMin(&smin[1], e1); atomicMax(&smax[1], e1);
    atomicMin(&smin[2], e2); atomicMax(&smax[2], e2);
    atomicMin(&smin[3], e3); atomicMax(&smax[3], e3);
  }
  __syncthreads();
  if (tid < 4) { atomicMin(&gmm[tid], smin[tid]); atomicMax(&gmm[4 + tid], smax[tid]); }
}

__global__ void k_params(const unsigned* gmm, Params* prm) {
  if (threadIdx.x != 0 || blockIdx.x != 0) return;
  long long ks = 1;
  for (int d = 0; d < 4; ++d) {
    float vs = (d == 0) ? 1.0f : 0.5f;
    float mn = fdec(gmm[d]);
    float mx = fdec(gmm[4 + d]);
    float pcmin = mn - 2.0f * vs;
    float pcmax = mx + 2.0f * vs;
    int dim = (int)rintf((pcmax - pcmin) / vs) + 3;  // round-half-even == jnp.round
    prm->pcmin[d] = pcmin;
    prm->dims[d]  = dim;
    ks *= dim;
  }
  prm->keyspace = (ks > K_CAP) ? K_CAP : (int)ks;
}

__global__ void k_clear(int* __restrict__ mark, int markN,
                        float* __restrict__ accum, int accN) {
  int i = blockIdx.x * blockDim.x + threadIdx.x;
  if (i < markN) mark[i] = 0;
  if (i < accN)  accum[i] = 0.0f;
}

__global__ void k_keys(const float* __restrict__ pts, int n,
                       const Params* __restrict__ prm,
                       int* __restrict__ pkeys, int* __restrict__ mark) {
  int i = blockIdx.x * blockDim.x + threadIdx.x;
  if (i >= n) return;
  float4 p = reinterpret_cast<const float4*>(pts)[i];
  int c0 = (int)rintf((p.x - prm->pcmin[0]) / 1.0f) + 1;
  int c1 = (int)rintf((p.y - prm->pcmin[1]) / 0.5f) + 1;
  int c2 = (int)rintf((p.z - prm->pcmin[2]) / 0.5f) + 1;
  int c3 = (int)rintf((p.w - prm->pcmin[3]) / 0.5f) + 1;
  int key = ((c0 * prm->dims[1] + c1) * prm->dims[2] + c2) * prm->dims[3] + c3;
  key = min(max(key, 0), K_CAP - 1);
  pkeys[i] = key;
  mark[key] = 1;   // all writers write 1: no atomic needed
}

// -------- exclusive prefix scan over the dense key space (3 passes) --------
__global__ void k_scan1(const int* __restrict__ mark, int* __restrict__ prefix,
                        int* __restrict__ bsum) {
  __shared__ int s[256];
  int tid = threadIdx.x;
#if HAS_ASYNC_LDS
  // stage the tile LDS-side with the gfx1250 async global->LDS path
  __shared__ int tile[SCAN_TILE];
  __builtin_amdgcn_global_load_async_to_lds_b128(
      (__attribute__((address_space(1))) v4i*)(mark + blockIdx.x * SCAN_TILE + tid * 4),
      (__attribute__((address_space(3))) v4i*)(&tile[tid * 4]), 0, 0);
  __builtin_amdgcn_s_wait_asynccnt(0);
  __syncthreads();
  int4 v = *reinterpret_cast<const int4*>(&tile[tid * 4]);
#else
  int4 v = reinterpret_cast<const int4*>(mark)[blockIdx.x * (SCAN_TILE / 4) + tid];
#endif
  int t = v.x + v.y + v.z + v.w;
  s[tid] = t;
  __syncthreads();
  #pragma unroll
  for (int off = 1; off < 256; off <<= 1) {
    int add = (tid >= off) ? s[tid - off] : 0;
    __syncthreads();
    s[tid] += add;
    __syncthreads();
  }
  int excl = s[tid] - t;
  if (tid == 255) bsum[blockIdx.x] = s[tid];
  int4 o;
  o.x = excl; o.y = excl + v.x; o.z = excl + v.x + v.y; o.w = excl + v.x + v.y + v.z;
  reinterpret_cast<int4*>(prefix)[blockIdx.x * (SCAN_TILE / 4) + tid] = o;
}

__global__ void k_scan2(int* __restrict__ bsum, Params* prm) {
  __shared__ int s[256];
  int tid = threadIdx.x;
  int4 v = reinterpret_cast<int4*>(bsum)[tid];   // SCAN_BLOCKS == 1024 entries
  int t = v.x + v.y + v.z + v.w;
  s[tid] = t;
  __syncthreads();
  #pragma unroll
  for (int off = 1; off < 256; off <<= 1) {
    int add = (tid >= off) ? s[tid - off] : 0;
    __syncthreads();
    s[tid] += add;
    __syncthreads();
  }
  int excl = s[tid] - t;
  int4 o;
  o.x = excl; o.y = excl + v.x; o.z = excl + v.x + v.y; o.w = excl + v.x + v.y + v.z;
  reinterpret_cast<int4*>(bsum)[tid] = o;
  if (tid == 255) prm->numUnique = excl + t;     // total #unique voxels
}

// fold mark+block-offset into a single "rank" array:
//   rank[k] = mark[k] ? global exclusive prefix (the unique-rank) : nDump
__global__ void k_scan3(int* __restrict__ prefix, const int* __restrict__ mark,
                        const int* __restrict__ bsum, int nDump) {
  int add = bsum[blockIdx.x];
  int idx = blockIdx.x * (SCAN_TILE / 4) + threadIdx.x;
  int4 m = reinterpret_cast<const int4*>(mark)[idx];
  int4 v = reinterpret_cast<int4*>(prefix)[idx];
  v.x = m.x ? v.x + add : nDump;
  v.y = m.y ? v.y + add : nDump;
  v.z = m.z ? v.z + add : nDump;
  v.w = m.w ? v.w + add : nDump;
  reinterpret_cast<int4*>(prefix)[idx] = v;
}

__global__ void k_group(const int* __restrict__ pkeys, const int* __restrict__ rank,
                        int n, float* __restrict__ outGid) {
  int i = blockIdx.x * blockDim.x + threadIdx.x;
  if (i >= n) return;
  outGid[i] = (float)rank[pkeys[i]];   // point keys are marked -> true rank
}

// -------- 27-stencil weighted accumulation: one wave per point -----------
// A(16x4): column K=0 holds 16 neighbor weights. B(4x16): row K=0 holds the
// 14-component feature vector f = [1, p4, outer(xyz)].  D = A*B gives all
// 16x14 scatter values in one V_WMMA_F32_16X16X4_F32.  The scatter is fully
// branch-free: invalid/padded rows go to the dump row (index n), and the two
// padded columns carry f=0 so their atomic adds are zeros.
__global__ void __launch_bounds__(256)
k_accum(const float* __restrict__ pts, int n, const Params* __restrict__ prm,
        const int* __restrict__ rank, float* __restrict__ accum) {
  const int lane = threadIdx.x & 31;
  const int i = blockIdx.x * 8 + (threadIdx.x >> 5);
  if (i >= n) return;                          // wave-uniform: EXEC stays all-ones

  float4 p = reinterpret_cast<const float4*>(pts)[i];
  __builtin_prefetch(pts + 4 * (i + 8), 0, 0);       // next block's row: global_prefetch

  const float pm0 = prm->pcmin[0], pm1 = prm->pcmin[1];
  const float pm2 = prm->pcmin[2], pm3 = prm->pcmin[3];
  const int d1 = prm->dims[1], d2 = prm->dims[2], d3 = prm->dims[3];

  const int c0 = (int)rintf((p.x - pm0) / 1.0f) + 1;
  const int c1 = (int)rintf((p.y - pm1) / 0.5f) + 1;
  const int c2 = (int)rintf((p.z - pm2) / 0.5f) + 1;
  const int c3 = (int)rintf((p.w - pm3) / 0.5f) + 1;

  // lane j < 27 owns stencil offset (meshgrid 'ij' order): cand = vcoor - off
  const int j  = lane;
  const int o1 = j / 9 - 1, o2 = (j / 3) % 3 - 1, o3 = j % 3 - 1;
  const int n1 = c1 - o1, n2 = c2 - o2, n3 = c3 - o3;
  int ckey = ((c0 * d1 + n1) * d2 + n2) * d3 + n3;
  ckey = min(max(ckey, 0), K_CAP - 1);
  const int  seg   = (j < 27) ? rank[ckey] : n;      // n == dump row
  const bool valid = (seg < n);

  const float cx = pm1 + (float)(n1 - 1) * 0.5f;
  const float cy = pm2 + (float)(n2 - 1) * 0.5f;
  const float cz = pm3 + (float)(n3 - 1) * 0.5f;
  const float dx = p.y - cx, dy = p.z - cy, dz = p.w - cz;
  // decay_radius^2 = 3*0.25/4 = 0.1875
  const float w = valid ? expf(-(dx*dx + dy*dy + dz*dz) * (1.0f / 0.1875f)) : 0.0f;

  // this lane's output column c = lane&15 of f (components 14,15 are zero)
  const int c = lane & 15;
  float fc;
  if (c == 0)      fc = 1.0f;
  else if (c <= 4) fc = (c == 1) ? p.x : ((c == 2) ? p.y : ((c == 3) ? p.z : p.w));
  else {
    int cc = c - 5;                                  // 0..10
    int ra = cc / 3, rb = cc - ra * 3;
    float fa = (ra == 0) ? p.y : ((ra == 1) ? p.z : p.w);
    float fb = (rb == 0) ? p.y : ((rb == 1) ? p.z : p.w);
    fc = (cc <= 8) ? fa * fb : 0.0f;
  }

#if HAS_WMMA_F32X4
  v2f bmat;                                          // B row K=0 = f, rest 0
  bmat.x = (lane < 16) ? fc : 0.0f;
  bmat.y = 0.0f;
#endif
  const float w16  = __shfl_xor(w, 16, 32);          // half-swap for rows 16..31
  const int   half = (lane >> 4) << 3;
  float* colp = accum + c;                           // this lane's column base

  #pragma unroll
  for (int j0 = 0; j0 < 32; j0 += 16) {
    float wrow = (j0 == 0) ? w : w16;                // lanes 0..15 carry A col K=0
#if HAS_WMMA_F32X4
    v2f amat; amat.x = (lane < 16) ? wrow : 0.0f; amat.y = 0.0f;
    v8f cacc = {0.f,0.f,0.f,0.f,0.f,0.f,0.f,0.f};
    v8f dmat = __builtin_amdgcn_wmma_f32_16x16x4_f32(
        false, amat, false, bmat, (short)0, cacc, false, false);
#else
    v8f dmat;
    #pragma unroll
    for (int r = 0; r < 8; ++r)
      dmat[r] = __shfl(w, j0 + r + half, 32) * fc;
#endif
    #pragma unroll
    for (int r = 0; r < 8; ++r) {
      // segment ids for the two half-rows as wave-uniform SGPRs (no LDS)
      int sLo = __builtin_amdgcn_readlane(seg, j0 + r);
      int sHi = __builtin_amdgcn_readlane(seg, j0 + r + 8);
      int sj  = (lane < 16) ? sLo : sHi;
      atomicAdd(&colp[(size_t)sj * NSTRIDE], dmat[r]);   // unconditional scatter
    }
  }
}

// ----------------------- 3x3 symmetric eigensolver -----------------------
__device__ __forceinline__ void cross3(const float* a, const float* b, float* o) {
  o[0] = a[1]*b[2] - a[2]*b[1];
  o[1] = a[2]*b[0] - a[0]*b[2];
  o[2] = a[0]*b[1] - a[1]*b[0];
}
__device__ __forceinline__ float dot3(const float* a, const float* b) {
  return a[0]*b[0] + a[1]*b[1] + a[2]*b[2];
}
__device__ __forceinline__ bool norm3(float* v) {
  float n2 = dot3(v, v);
  if (n2 < 1e-24f) return false;
  float inv = rsqrtf(n2);
  v[0] *= inv; v[1] *= inv; v[2] *= inv;
  return true;
}
__device__ __forceinline__ void orth3(const float* u, float* v) {
  float ax = fabsf(u[0]), ay = fabsf(u[1]), az = fabsf(u[2]);
  float e[3] = {0.f, 0.f, 0.f};
  if (ax <= ay && ax <= az) e[0] = 1.f;
  else if (ay <= az)        e[1] = 1.f;
  else                      e[2] = 1.f;
  cross3(u, e, v);
  norm3(v);
}
__device__ bool eigvec(float a00, float a01, float a02, float a11, float a12,
                       float a22, float lam, float* v) {
  float r0[3] = {a00 - lam, a01, a02};
  float r1[3] = {a01, a11 - lam, a12};
  float r2[3] = {a02, a12, a22 - lam};
  float c01[3], c02[3], c12[3];
  cross3(r0, r1, c01); cross3(r0, r2, c02); cross3(r1, r2, c12);
  float n01 = dot3(c01, c01), n02 = dot3(c02, c02), n12 = dot3(c12, c12);
  float nb = n01;
  v[0] = c01[0]; v[1] = c01[1]; v[2] = c01[2];
  if (n02 > nb) { nb = n02; v[0] = c02[0]; v[1] = c02[1]; v[2] = c02[2]; }
  if (n12 > nb) { nb = n12; v[0] = c12[0]; v[1] = c12[1]; v[2] = c12[2]; }
  float rn = dot3(r0, r0) + dot3(r1, r1) + dot3(r2, r2);
  if (nb <= 1e-12f * rn * rn + 1e-30f) return false;
  return norm3(v);
}

__global__ void k_final(const float* __restrict__ accum, const Params* __restrict__ prm,
                        int n, float* __restrict__ outMu, float* __restrict__ outR) {
  int u = blockIdx.x * blockDim.x + threadIdx.x;
  if (u >= n) return;
  int nu = prm->numUnique;
  if (u >= nu) {                      // padded rows: mu=0, R=1e-6*I (matches ref)
    #pragma unroll
    for (int k = 0; k < 4; ++k) outMu[(size_t)u*4 + k] = 0.0f;
    #pragma unroll
    for (int k = 0; k < 9; ++k) outR[(size_t)u*9 + k] = (k % 4 == 0) ? 1e-6f : 0.0f;
    return;
  }
  float s[NCOMP];
  #pragma unroll
  for (int k = 0; k < NCOMP; ++k) s[k] = accum[(size_t)u * NSTRIDE + k];
  float denom = fmaxf(s[0], 1e-12f);
  float inv = 1.0f / denom;
  float mu[4];
  #pragma unroll
  for (int k = 0; k < 4; ++k) { mu[k] = s[1 + k] * inv; outMu[(size_t)u*4 + k] = mu[k]; }

  // symmetric covariance
  float m00 = s[5]  * inv - mu[1] * mu[1];
  float m11 = s[9]  * inv - mu[2] * mu[2];
  float m22 = s[13] * inv - mu[3] * mu[3];
  float m01 = 0.5f * (s[6]  + s[8])  * inv - mu[1] * mu[2];
  float m02 = 0.5f * (s[7]  + s[11]) * inv - mu[1] * mu[3];
  float m12 = 0.5f * (s[10] + s[12]) * inv - mu[2] * mu[3];

  float p1q = m01*m01 + m02*m02 + m12*m12;
  float q   = (m00 + m11 + m22) * (1.0f / 3.0f);
  float b00 = m00 - q, b11 = m11 - q, b22 = m22 - q;
  float p2q = b00*b00 + b11*b11 + b22*b22 + 2.0f * p1q;

  float l0, l1, l2;
  float V[3][3];
  bool identity = false;
  if (p2q <= 1e-18f) {
    l0 = l1 = l2 = q;
    identity = true;
  } else {
    float pp  = sqrtf(p2q * (1.0f / 6.0f));
    float det = b00 * (b11 * b22 - m12 * m12)
              - m01 * (m01 * b22 - m12 * m02)
              + m02 * (m01 * m12 - b11 * m02);
    float r = det / (2.0f * pp * pp * pp);
    r = fminf(1.0f, fmaxf(-1.0f, r));
    float phi = acosf(r) * (1.0f / 3.0f);
    l0 = q + 2.0f * pp * cosf(phi);
    l2 = q + 2.0f * pp * cosf(phi + 2.0943951023931953f);  // +2pi/3 -> smallest
    l1 = 3.0f * q - l0 - l2;

    float v0[3], v2[3];
    bool ok0 = eigvec(m00, m01, m02, m11, m12, m22, l0, v0);
    bool ok2 = eigvec(m00, m01, m02, m11, m12, m22, l2, v2);
    if (ok0 && ok2) {
      float d = dot3(v2, v0);                 // re-orthogonalize
      v2[0] -= d * v0[0]; v2[1] -= d * v0[1]; v2[2] -= d * v0[2];
      if (!norm3(v2)) ok2 = false;
    }
    if (ok0 && !ok2)      orth3(v0, v2);
    else if (!ok0 && ok2) orth3(v2, v0);
    else if (!ok0 && !ok2) identity = true;

    if (!identity) {
      float v1[3];
      cross3(v2, v0, v1);                     // columns (v0, v1, v2): det = +1
      #pragma unroll
      for (int r2i = 0; r2i < 3; ++r2i) {
        V[r2i][0] = v0[r2i]; V[r2i][1] = v1[r2i]; V[r2i][2] = v2[r2i];
      }
    }
  }
  if (identity) {
    #pragma unroll
    for (int a = 0; a < 3; ++a)
      #pragma unroll
      for (int b = 0; b < 3; ++b) V[a][b] = (a == b) ? 1.0f : 0.0f;
  }
  float S0 = fmaxf(sqrtf(fmaxf(l0, 0.0f)), 1e-6f);
  float S1 = fmaxf(sqrtf(fmaxf(l1, 0.0f)), 1e-6f);
  float S2 = fmaxf(sqrtf(fmaxf(l2, 0.0f)), 1e-6f);
  #pragma unroll
  for (int a = 0; a < 3; ++a) {
    outR[(size_t)u*9 + a*3 + 0] = V[a][0] * S0;
    outR[(size_t)u*9 + a*3 + 1] = V[a][1] * S1;
    outR[(size_t)u*9 + a*3 + 2] = V[a][2] * S2;
  }
}

// ---------------------------------------------------------------------------
extern "C" void kernel_launch(void* const* d_in, const int* in_sizes, int n_in,
                              void* d_out, int out_size, void* d_ws, size_t ws_size,
                              hipStream_t stream) {
  (void)n_in; (void)out_size; (void)ws_size;
  const float* pts = (const float*)d_in[0];
  const int n = in_sizes[0] / 4;          // 200000

  float* out   = (float*)d_out;
  float* outMu = out;                      // [n,4]
  float* outR  = out + (size_t)4 * n;      // [n,9]
  float* outG  = out + (size_t)13 * n;     // [n]   group_ids as float

  // workspace layout
  char*     ws     = (char*)d_ws;
  unsigned* gmm    = (unsigned*)ws;                     // 8 u32
  Params*   prm    = (Params*)(ws + 64);
  int*      mark   = (int*)(ws + 128);                  // K_CAP ints
  int*      prefix = mark + K_CAP;                      // K_CAP ints (becomes rank)
  int*      bsum   = prefix + K_CAP;                    // SCAN_BLOCKS ints
  int*      pkeys  = bsum + SCAN_BLOCKS;                // n ints
  float*    accum  = (float*)(pkeys + ((n + 3) & ~3)); // (n+1)*NSTRIDE floats

  const int B  = 256;
  const int gN = (n + B - 1) / B;
  const int accN   = (n + 1) * NSTRIDE;
  const int clearN = (K_CAP > accN) ? K_CAP : accN;

  k_init  <<<1, 32, 0, stream>>>(gmm);
  k_minmax<<<gN, B, 0, stream>>>(pts, n, gmm);
  k_params<<<1, 1, 0, stream>>>(gmm, prm);
  k_clear <<<(clearN + B - 1) / B, B, 0, stream>>>(mark, K_CAP, accum, accN);
  k_keys  <<<gN, B, 0, stream>>>(pts, n, prm, pkeys, mark);
  k_scan1 <<<SCAN_BLOCKS, 256, 0, stream>>>(mark, prefix, bsum);
  k_scan2 <<<1, 256, 0, stream>>>(bsum, prm);
  k_scan3 <<<SCAN_BLOCKS, 256, 0, stream>>>(prefix, mark, bsum, n);
  k_group <<<gN, B, 0, stream>>>(pkeys, prefix, n, outG);
  k_accum <<<(n + 7) / 8, 256, 0, stream>>>(pts, n, prm, prefix, accum);
  k_final <<<gN, B, 0, stream>>>(accum, prm, n, outMu, outR);
}